// gnn_backbone_28329604284662
// MI455X (gfx1250) — compile-verified
//
#include <hip/hip_runtime.h>

// GNN backbone: 2x [ TAGConv(K=2, D=32) -> BatchNorm(train) -> LeakyReLU(0.01) ]
// N=100000 nodes, E=1.6M edges, D=32.
//
// Design notes (MI455X):
//  * Node feature buffer = N*32*4B = 12.8 MB -> L2-resident (192 MB), so the
//    edge gather/scatter passes are L2-bound, not HBM-bound. Use native
//    global_atomic_add_f32 (unsafeAtomicAdd), one wave per edge, lane=feature.
//  * GEMMs are [N,32]@[32,32]: run on the matrix pipe with
//    V_WMMA_F32_16X16X4_F32. One wave owns 16 rows x 32 cols = two 16x16
//    f32 accumulators, 8 K-steps of 4. N/16 = 6250 tiles exactly.
//  * TAGConv bias cancels exactly under training-mode BatchNorm (constant
//    per-column shift is removed by the batch mean, variance unchanged).

#define D 32

typedef __attribute__((ext_vector_type(2))) float v2f;
typedef __attribute__((ext_vector_type(8))) float v8f;

__global__ void zero_f32(float* __restrict__ p, int n) {
  int i = blockIdx.x * blockDim.x + threadIdx.x;
  int stride = gridDim.x * blockDim.x;
  for (; i < n; i += stride) p[i] = 0.0f;
}

// Out[16 rows x 32 cols per wave] (+)= X[16x32] @ W[32x32]
// A-fragment (32-bit 16x4): (v0,lanes0-15)=K0, (v1,lo)=K1, (v0,hi)=K2, (v1,hi)=K3
// C/D (f32 16x16): VGPR r <-> M = r + 8*(lane>=16), N = lane%16
__global__ void gemm32_wmma(const float* __restrict__ X, const float* __restrict__ W,
                            float* __restrict__ Out, int nTiles, int accumulate) {
  int wave = (int)((blockIdx.x * blockDim.x + threadIdx.x) >> 5);
  if (wave >= nTiles) return;               // wave-uniform branch, EXEC stays all-1s
  int lane = threadIdx.x & 31;
  int l15  = lane & 15;
  int hi   = lane >> 4;                     // 0: lanes 0-15, 1: lanes 16-31
  const float* xrow = X + (size_t)(wave * 16 + l15) * D;

  v8f c0 = {}, c1 = {};
  if (accumulate) {
#pragma unroll
    for (int r = 0; r < 8; ++r) {
      int m = wave * 16 + r + 8 * hi;
      c0[r] = Out[(size_t)m * D + l15];
      c1[r] = Out[(size_t)m * D + 16 + l15];
    }
  }

#pragma unroll
  for (int kk = 0; kk < 8; ++kk) {
    int k0 = kk * 4 + hi * 2;               // this half-wave's K pair
    v2f a, b0, b1;
    a.x  = xrow[k0];
    a.y  = xrow[k0 + 1];
    b0.x = W[(size_t)k0 * D + l15];
    b0.y = W[(size_t)(k0 + 1) * D + l15];
    b1.x = W[(size_t)k0 * D + 16 + l15];
    b1.y = W[(size_t)(k0 + 1) * D + 16 + l15];
    // (neg_a, A, neg_b, B, c_mod, C, reuse_a, reuse_b)
    c0 = __builtin_amdgcn_wmma_f32_16x16x4_f32(false, a, false, b0, (short)0, c0, false, false);
    c1 = __builtin_amdgcn_wmma_f32_16x16x4_f32(false, a, false, b1, (short)0, c1, false, false);
  }

#pragma unroll
  for (int r = 0; r < 8; ++r) {
    int m = wave * 16 + r + 8 * hi;
    Out[(size_t)m * D + l15]      = c0[r];
    Out[(size_t)m * D + 16 + l15] = c1[r];
  }
}

// Weighted scatter-add propagation: Hn[dst] += H[src] * w.
// One wave per edge, lane = feature column -> coalesced 128B row accesses,
// native fp32 global atomics into an L2-resident 12.8 MB buffer.
__global__ void propagate(const float* __restrict__ H, float* __restrict__ Hn,
                          const int* __restrict__ src, const int* __restrict__ dst,
                          const float* __restrict__ ewt, int nE) {
  int lane = threadIdx.x & 31;
  int wave = (int)((blockIdx.x * blockDim.x + threadIdx.x) >> 5);
  int nWaves = (int)((gridDim.x * blockDim.x) >> 5);
  for (int e = wave; e < nE; e += nWaves) {
    int s = src[e];
    int d = dst[e];
    float w = ewt[e];
    float v = H[(size_t)s * D + lane] * w;
    unsafeAtomicAdd(&Hn[(size_t)d * D + lane], v);
  }
}

// Per-column sum and sum-of-squares -> stats[0..31], stats[32..63]
__global__ void bn_stats(const float* __restrict__ X, float* __restrict__ stats, int n) {
  int col = threadIdx.x & 31;
  int rowInBlock = threadIdx.x >> 5;                          // 0..7
  int rowsPerGrid = (int)(gridDim.x * (blockDim.x >> 5));
  float s = 0.0f, s2 = 0.0f;
  for (int r = (int)(blockIdx.x * (blockDim.x >> 5)) + rowInBlock; r < n; r += rowsPerGrid) {
    float v = X[(size_t)r * D + col];
    s += v;
    s2 += v * v;
  }
  __shared__ float ls[8][D];
  __shared__ float ls2[8][D];
  ls[rowInBlock][col]  = s;
  ls2[rowInBlock][col] = s2;
  __syncthreads();
  if (rowInBlock == 0) {
#pragma unroll
    for (int i = 1; i < 8; ++i) {
      s  += ls[i][col];
      s2 += ls2[i][col];
    }
    unsafeAtomicAdd(&stats[col], s);
    unsafeAtomicAdd(&stats[D + col], s2);
  }
}

// stats -> per-column scale/shift: scale = gamma*rsqrt(var+eps), shift = beta - mean*scale
__global__ void bn_finalize(float* __restrict__ stats, const float* __restrict__ gamma,
                            const float* __restrict__ beta, float inv_n) {
  int c = threadIdx.x;
  float m  = stats[c] * inv_n;
  float v  = stats[D + c] * inv_n - m * m;   // biased variance
  float sc = gamma[c] * rsqrtf(v + 1e-5f);
  stats[2 * D + c] = sc;
  stats[3 * D + c] = beta[c] - m * sc;
}

// Fused normalize + LeakyReLU(0.01)
__global__ void bn_apply(const float* __restrict__ X, const float* __restrict__ stats,
                         float* __restrict__ Y, int total) {
  int col = threadIdx.x & 31;                // valid: blockDim and stride are multiples of 32
  float sc = stats[2 * D + col];
  float sh = stats[3 * D + col];
  int i = blockIdx.x * blockDim.x + threadIdx.x;
  int stride = gridDim.x * blockDim.x;
  for (; i < total; i += stride) {
    float v = X[i] * sc + sh;
    Y[i] = v > 0.0f ? v : 0.01f * v;
  }
}

extern "C" void kernel_launch(void* const* d_in, const int* in_sizes, int n_in,
                              void* d_out, int out_size, void* d_ws, size_t ws_size,
                              hipStream_t stream) {
  const float* y      = (const float*)d_in[0];
  const int*   ei     = (const int*)d_in[1];
  const float* ew     = (const float*)d_in[2];
  const float* W1     = (const float*)d_in[3];
  // d_in[4] = bias1 : cancels exactly under training-mode BN (skipped)
  const float* gamma1 = (const float*)d_in[5];
  const float* beta1  = (const float*)d_in[6];
  const float* W2     = (const float*)d_in[7];
  // d_in[8] = bias2 : cancels (skipped)
  const float* gamma2 = (const float*)d_in[9];
  const float* beta2  = (const float*)d_in[10];

  const int N = in_sizes[0] / D;        // 100000
  const int E = in_sizes[2];            // 1600000
  const int* src = ei;                  // edge_index[0]
  const int* dst = ei + E;              // edge_index[1]

  // Workspace: 3 node buffers (N*D f32 each) + 128-float stats block (~38.4 MB)
  float* buf0  = (float*)d_ws;                 // h scratch / layer-1 output
  float* buf1  = buf0 + (size_t)N * D;         // h scratch
  float* outb  = buf1 + (size_t)N * D;         // GEMM accumulator
  float* stats = outb + (size_t)N * D;         // 4*D floats

  const int nTiles = N / 16;                   // 6250 (exact)
  const int gemmBlocks = (nTiles + 3) / 4;     // 4 waves / block (128 threads)
  const int ND = N * D;
  const float inv_n = 1.0f / (float)N;

  // ---- per layer sequence ----
  auto run_layer = [&](const float* x, const float* W, const float* gamma,
                       const float* beta, float* out_final) {
    // hop 0: outb = x @ W[0]
    gemm32_wmma<<<gemmBlocks, 128, 0, stream>>>(x, W, outb, nTiles, 0);
    // hop 1: buf1 = A_w x ; outb += buf1 @ W[1]
    zero_f32<<<2048, 256, 0, stream>>>(buf1, ND);
    propagate<<<2048, 256, 0, stream>>>(x, buf1, src, dst, ew, E);
    gemm32_wmma<<<gemmBlocks, 128, 0, stream>>>(buf1, W + D * D, outb, nTiles, 1);
    // hop 2: buf0 = A_w buf1 ; outb += buf0 @ W[2]   (x fully consumed by now)
    zero_f32<<<2048, 256, 0, stream>>>(buf0, ND);
    propagate<<<2048, 256, 0, stream>>>(buf1, buf0, src, dst, ew, E);
    gemm32_wmma<<<gemmBlocks, 128, 0, stream>>>(buf0, W + 2 * D * D, outb, nTiles, 1);
    // BatchNorm (training stats) + LeakyReLU, fused apply
    zero_f32<<<1, 64, 0, stream>>>(stats, 2 * D);
    bn_stats<<<1024, 256, 0, stream>>>(outb, stats, N);
    bn_finalize<<<1, D, 0, stream>>>(stats, gamma, beta, inv_n);
    bn_apply<<<2048, 256, 0, stream>>>(outb, stats, out_final, ND);
  };

  run_layer(y,    W1, gamma1, beta1, buf0);          // layer 1 -> buf0
  run_layer(buf0, W2, gamma2, beta2, (float*)d_out); // layer 2 -> d_out
}